// Rotational_12232066859560
// MI455X (gfx1250) — compile-verified
//
#include <hip/hip_runtime.h>
#include <hip/hip_bf16.h>

// Problem constants (from reference: inp (8,1024,4096) f32, 2048 pairs)
constexpr int N       = 4096;            // nodes (columns, contiguous)
constexpr int NPAIRS  = N / 2;           // 2048 rotation pairs
constexpr int TPB     = 256;             // threads per block (8 waves on wave32)
constexpr int PPT     = NPAIRS / TPB;    // 8 pairs per thread
constexpr int V4      = N / 4;           // 1024 float4 per row
constexpr int V4PT    = V4 / TPB;        // 4 float4 per thread per row
constexpr int RPB     = 8;               // rows per block (amortize metadata, long pipeline)

// Native vector types (clang builtins reject HIP_vector_type classes)
typedef int   v4i __attribute__((__vector_size__(4 * sizeof(int))));
typedef float v4f __attribute__((__vector_size__(4 * sizeof(float))));

#if defined(__HIP_DEVICE_COMPILE__) && \
    __has_builtin(__builtin_amdgcn_global_load_async_to_lds_b128) && \
    __has_builtin(__builtin_amdgcn_s_wait_asynccnt)
#define USE_ASYNC_LDS 1
#else
#define USE_ASYNC_LDS 0
#endif

// ---------------------------------------------------------------------------
// Prep: pack (i0,i1,o0,o1) and (cos,sin) per pair into workspace. 48 KB total.
// ---------------------------------------------------------------------------
__global__ __launch_bounds__(TPB) void rot_prep(
    const float* __restrict__ angles,
    const int*   __restrict__ pairs,
    const int*   __restrict__ outp_pairs,
    int4*        __restrict__ widx,
    float2*      __restrict__ wcs) {
  const int k = blockIdx.x * TPB + threadIdx.x;
  if (k < NPAIRS) {
    int4 v;
    v.x = pairs[2 * k];
    v.y = pairs[2 * k + 1];
    v.z = outp_pairs[2 * k];
    v.w = outp_pairs[2 * k + 1];
    widx[k] = v;
    const float a = angles[k];
    wcs[k] = make_float2(cosf(a), sinf(a));
  }
}

// ---------------------------------------------------------------------------
// Main: per block, 8 rows, software-pipelined. Async-DMA row r+1 into the
// alternate LDS buffer (ASYNCcnt in-order completion => s_wait_asynccnt(4)
// guarantees row r has landed), rotate/permute row r inside LDS, stream it
// out with coalesced non-temporal b128 stores.
// ---------------------------------------------------------------------------
__global__ __launch_bounds__(TPB) void rot_main(
    const float* __restrict__ inp,
    const int4*  __restrict__ widx,
    const float2* __restrict__ wcs,
    float*       __restrict__ outp) {
  __shared__ __align__(16) float sIn[2][N];   // double-buffered input rows
  __shared__ __align__(16) float sOut[N];     // permuted output row

  const int  tid  = threadIdx.x;
  const long row0 = (long)blockIdx.x * RPB;

  // Issue staging of row r into LDS buffer b (coalesced b128 path).
  auto stage = [&](int r, int b) {
    const v4f* __restrict__ src4 = (const v4f*)(inp + (size_t)(row0 + r) * N);
    v4f* buf4 = (v4f*)sIn[b];
#if USE_ASYNC_LDS
#pragma unroll
    for (int v = 0; v < V4PT; ++v) {
      const int e = v * TPB + tid;
      __builtin_amdgcn_global_load_async_to_lds_b128(
          (__attribute__((address_space(1))) v4i*)(src4 + e),
          (__attribute__((address_space(3))) v4i*)(buf4 + e),
          /*offset=*/0, /*cpol=*/0);
    }
#else
#pragma unroll
    for (int v = 0; v < V4PT; ++v) {
      const int e = v * TPB + tid;
      buf4[e] = __builtin_nontemporal_load(src4 + e);
    }
#endif
  };

  // Kick off row 0 DMA first, then load per-thread pair metadata (overlaps).
  stage(0, 0);

  int4   idx[PPT];
  float2 cs[PPT];
#pragma unroll
  for (int u = 0; u < PPT; ++u) {
    idx[u] = widx[u * TPB + tid];
    cs[u]  = wcs [u * TPB + tid];
  }

#pragma unroll
  for (int r = 0; r < RPB; ++r) {
    // Prefetch next row into the alternate buffer, then wait only for the
    // CURRENT row's 4 async copies (in-order completion per wave).
    if (r + 1 < RPB) {
      stage(r + 1, (r + 1) & 1);
#if USE_ASYNC_LDS
      __builtin_amdgcn_s_wait_asynccnt(V4PT);
#endif
    } else {
#if USE_ASYNC_LDS
      __builtin_amdgcn_s_wait_asynccnt(0);
#endif
    }
    __syncthreads();   // row r visible to all waves; prior sOut reads done

    // ---- gather / rotate / scatter entirely in LDS ----
    const float* __restrict__ in = sIn[r & 1];
#pragma unroll
    for (int u = 0; u < PPT; ++u) {
      const float xi = in[idx[u].x];
      const float xj = in[idx[u].y];
      const float c  = cs[u].x;
      const float s  = cs[u].y;
      sOut[idx[u].z] = fmaf(c, xi, -s * xj);   // yi = c*xi - s*xj
      sOut[idx[u].w] = fmaf(c, xj,  s * xi);   // yj = c*xj + s*xi
    }
    __syncthreads();   // sOut complete; sIn[r&1] reads complete

    // ---- stream row out (coalesced, non-temporal: written once, never read) ----
    v4f* __restrict__ dst4 = (v4f*)(outp + (size_t)(row0 + r) * N);
#pragma unroll
    for (int v = 0; v < V4PT; ++v) {
      const int e = v * TPB + tid;
      const v4f val = ((const v4f*)sOut)[e];
      __builtin_nontemporal_store(val, dst4 + e);
    }
  }
}

// ---------------------------------------------------------------------------
extern "C" void kernel_launch(void* const* d_in, const int* in_sizes, int n_in,
                              void* d_out, int out_size, void* d_ws, size_t ws_size,
                              hipStream_t stream) {
  const float* inp        = (const float*)d_in[0];
  const float* angles     = (const float*)d_in[1];
  const int*   pairs      = (const int*)d_in[2];
  const int*   outp_pairs = (const int*)d_in[3];
  float*       out        = (float*)d_out;

  int4*   widx = (int4*)d_ws;
  float2* wcs  = (float2*)((char*)d_ws + (size_t)NPAIRS * sizeof(int4));

  rot_prep<<<NPAIRS / TPB, TPB, 0, stream>>>(angles, pairs, outp_pairs, widx, wcs);

  const int rows = out_size / N;            // 8192
  rot_main<<<rows / RPB, TPB, 0, stream>>>(inp, widx, wcs, out);
}